// BsplineEncoding_72851235274859
// MI455X (gfx1250) — compile-verified
//
#include <hip/hip_runtime.h>
#include <stdint.h>

// B-spline encoding, bandwidth-bound scatter.
// Per point layout: [x0, feat0[64], x1, feat1[64], x2, feat2[64]] -> 195 f32.

#define DIMS        3
#define KBINS       64
#define SEG         (1 + KBINS)            // 65 floats per (point,dim)
#define ROW         (DIMS * SEG)           // 195 floats per point
#define PTS_PER_BLK 64
#define TILE_FLOATS (PTS_PER_BLK * ROW)    // 12480 floats = 49920 bytes (16B multiple)
#define TILE_V4     (TILE_FLOATS / 4)      // 3120 float4s
#define THREADS     256
#define FULL_IT     (TILE_V4 / THREADS)    // 12 uniform iterations
#define REM         (TILE_V4 - FULL_IT * THREADS)  // 48 remainder lanes

typedef __attribute__((ext_vector_type(4))) float v4f;

__global__ __launch_bounds__(THREADS)
void bspline_encode_kernel(const float* __restrict__ x,
                           float* __restrict__ out,
                           int n_points)
{
    __shared__ v4f tile4[TILE_V4];                 // 16B-aligned LDS tile
    float* tile = (float*)tile4;

    const int tid = threadIdx.x;
    const int p0  = blockIdx.x * PTS_PER_BLK;      // first point of this tile

    // ---- Phase 1: zero-fill LDS tile, uniform fully-unrolled ds_store_b128 ----
    const v4f z = {0.0f, 0.0f, 0.0f, 0.0f};
    #pragma unroll
    for (int i = 0; i < FULL_IT; ++i)
        tile4[tid + i * THREADS] = z;
    if (tid < REM)
        tile4[tid + FULL_IT * THREADS] = z;
    __syncthreads();

    // ---- Phase 2: scatter x + 4 cubic B-spline weights per (point,dim) ----
    // tid enumerates (p,d) pairs; segment offset = p*195 + d*65 = 65*tid,
    // and the input gather x[(p0+p)*3 + d] = x[p0*3 + tid] is fully coalesced.
    if (tid < PTS_PER_BLK * DIMS) {
        const int n3 = p0 * DIMS + tid;
        if (n3 < n_points * DIMS) {
            const float xv = x[n3];
            // SCALE = (K - DEG)/(MAX - MIN) = 61/2 = 30.5 ; f32(61-1e-6)==61.0f
            float xs = (xv + 1.0f) * 30.5f;
            xs = fminf(fmaxf(xs, 0.0f), 61.0f);
            int idx = (int)xs;                       // xs >= 0 -> trunc == floor
            idx = (idx > KBINS - 4) ? (KBINS - 4) : idx;   // clamp to 60
            const float u   = xs - (float)idx;       // may be exactly 1.0 at edge
            const float u2  = u * u;
            const float u3  = u2 * u;
            const float omu = 1.0f - u;
            const float S6  = 1.0f / 6.0f;
            const float c0  = omu * omu * omu * S6;
            const float c1  = (3.0f * u3 - 6.0f * u2 + 4.0f) * S6;
            const float c2  = (-3.0f * u3 + 3.0f * u2 + 3.0f * u + 1.0f) * S6;
            const float c3  = u3 * S6;

            float* row = tile + tid * SEG;           // == p*ROW + d*SEG
            row[0]       = xv;
            row[1 + idx] = c0;
            row[2 + idx] = c1;
            row[3 + idx] = c2;
            row[4 + idx] = c3;
        }
    }
    __syncthreads();

    // ---- Phase 3: stream tile -> global ----
    const size_t base = (size_t)p0 * ROW;            // *4 bytes: 16B aligned
    if (p0 + PTS_PER_BLK <= n_points) {
        // CDNA5 async data mover: LDS -> global with no VGPR round-trip.
        // INST_OFFSET is added to BOTH the global and LDS addresses (ISA ch10.7),
        // so one voffset VGPR + one LDS-addr VGPR serve all 12 iterations.
        float* outp = out + base;                    // uniform -> SGPR pair
        const unsigned voff = (unsigned)tid * 16u;   // byte offset in tile
        const unsigned vlds = (unsigned)(uintptr_t)tile4 + voff; // LDS byte addr
        #pragma unroll
        for (int i = 0; i < FULL_IT; ++i) {
            asm volatile(
                "global_store_async_from_lds_b128 %0, %1, %2 offset:%3 th:TH_STORE_NT"
                :: "v"(voff), "v"(vlds), "s"(outp), "i"(i * THREADS * 16)
                : "memory");
        }
        if (tid < REM) {
            asm volatile(
                "global_store_async_from_lds_b128 %0, %1, %2 offset:%3 th:TH_STORE_NT"
                :: "v"(voff), "v"(vlds), "s"(outp), "i"(FULL_IT * THREADS * 16)
                : "memory");
        }
        // ASYNCcnt drains via the implicit wait-idle at s_endpgm.
    } else {
        // partial tail tile (not hit for N = 1,000,000, kept for generality)
        const int nf = (n_points - p0) * ROW;
        for (int i = tid; i < nf; i += THREADS)
            out[base + i] = tile[i];
    }
}

extern "C" void kernel_launch(void* const* d_in, const int* in_sizes, int n_in,
                              void* d_out, int out_size, void* d_ws, size_t ws_size,
                              hipStream_t stream) {
    (void)n_in; (void)out_size; (void)d_ws; (void)ws_size;
    const float* x  = (const float*)d_in[0];
    float* out      = (float*)d_out;
    const int n_pts = in_sizes[0] / DIMS;            // 1,000,000

    const int grid = (n_pts + PTS_PER_BLK - 1) / PTS_PER_BLK;   // 15625 blocks
    bspline_encode_kernel<<<grid, THREADS, 0, stream>>>(x, out, n_pts);
}